// GNNModel_37357625541229
// MI455X (gfx1250) — compile-verified
//
#include <hip/hip_runtime.h>
#include <math.h>

typedef float v2f __attribute__((ext_vector_type(2)));
typedef float v8f __attribute__((ext_vector_type(8)));

// ---------------------------------------------------------------------------
// Utility: zero a float buffer
// ---------------------------------------------------------------------------
__global__ void zero_f32(float* __restrict__ p, long n) {
  long i = (long)blockIdx.x * blockDim.x + threadIdx.x;
  if (i < n) p[i] = 0.0f;
}

// ---------------------------------------------------------------------------
// Degree count (mean aggregation denominator)
// ---------------------------------------------------------------------------
__global__ void degree_kernel(const int* __restrict__ dst, float* __restrict__ deg, int E) {
  int e = blockIdx.x * blockDim.x + threadIdx.x;
  if (e < E) atomicAdd(&deg[dst[e]], 1.0f);
}

// ---------------------------------------------------------------------------
// XW[n][k][o] = sum_i h[n][i] * W[k][i][o], in=16, out=16, 25 kernels.
// One wave per 16-node tile. A tile loaded once into registers, reused for
// all 25 k. Each k = 4 chained V_WMMA_F32_16X16X4_F32 (K accumulated 4 at a
// time). EXEC is all-1s at every WMMA; the store guard is wave-uniform so the
// common (full-tile) path is 8 unconditional stores with immediate offsets.
//
// A layout (16x4 f32): lane<16 holds row=lane, K={k0,k0+1}; lane>=16 holds
// row=lane-16, K={k0+2,k0+3}.  B layout (4x16) mirrors by column.
// C layout (16x16 f32, 8 VGPRs): row = r + 8*half, col = lane&15.
// ---------------------------------------------------------------------------
__global__ void xw16_wmma(const float* __restrict__ h,   // [N][16]
                          const float* __restrict__ W,   // [25][16][16]
                          float* __restrict__ XW,        // [N][25][16]
                          int N) {
  const int lane  = threadIdx.x & 31;
  const int wid   = threadIdx.x >> 5;
  const int ntile = (N + 15) >> 4;
  const int tile  = blockIdx.x * (blockDim.x >> 5) + wid;
  if (tile >= ntile) return;                 // wave-uniform exit
  const int half = lane >> 4;                // 0: low 16 lanes, 1: high
  const int lrow = lane & 15;
  const bool full = (tile * 16 + 16 <= N);   // wave-uniform
  int mrow = tile * 16 + lrow;
  if (mrow >= N) mrow = N - 1;               // branchless clamp for loads

  v2f a[4];
#pragma unroll
  for (int c = 0; c < 4; ++c) {
    const int kk = 4 * c + 2 * half;
    a[c].x = h[(long)mrow * 16 + kk + 0];
    a[c].y = h[(long)mrow * 16 + kk + 1];
  }

  for (int k = 0; k < 25; ++k) {
    v8f acc = {};
#pragma unroll
    for (int c = 0; c < 4; ++c) {
      const int kk = 4 * c + 2 * half;
      v2f b;
      b.x = W[k * 256 + (kk + 0) * 16 + lrow];
      b.y = W[k * 256 + (kk + 1) * 16 + lrow];
      acc = __builtin_amdgcn_wmma_f32_16x16x4_f32(
          /*neg_a=*/false, a[c], /*neg_b=*/false, b,
          /*c_mod=*/(short)0, acc, /*reuse_a=*/false, /*reuse_b=*/false);
    }
    if (full) {
#pragma unroll
      for (int r = 0; r < 8; ++r)
        XW[(long)(tile * 16 + r + 8 * half) * 400 + k * 16 + lrow] = acc[r];
    } else {
#pragma unroll
      for (int r = 0; r < 8; ++r) {
        const int row = tile * 16 + r + 8 * half;
        if (row < N) XW[(long)row * 400 + k * 16 + lrow] = acc[r];
      }
    }
  }
}

// ---------------------------------------------------------------------------
// Layer-1 variant: in=2 (x is [N][2], W1 is [25][2][16]).
// Single K=4 WMMA per kernel with the K>=2 half zero-masked (no divergence).
// ---------------------------------------------------------------------------
__global__ void xw2_wmma(const float* __restrict__ x,    // [N][2]
                         const float* __restrict__ W,    // [25][2][16]
                         float* __restrict__ XW,         // [N][25][16]
                         int N) {
  const int lane  = threadIdx.x & 31;
  const int wid   = threadIdx.x >> 5;
  const int ntile = (N + 15) >> 4;
  const int tile  = blockIdx.x * (blockDim.x >> 5) + wid;
  if (tile >= ntile) return;
  const int half = lane >> 4;
  const int lrow = lane & 15;
  const bool full = (tile * 16 + 16 <= N);        // wave-uniform
  const float mask = (half == 0) ? 1.0f : 0.0f;   // zero-pad K=2..3
  int mrow = tile * 16 + lrow;
  if (mrow >= N) mrow = N - 1;

  v2f a;
  a.x = x[(long)mrow * 2 + 0] * mask;
  a.y = x[(long)mrow * 2 + 1] * mask;

  for (int k = 0; k < 25; ++k) {
    v2f b;
    b.x = W[k * 32 + 0 * 16 + lrow] * mask;   // B row K=0
    b.y = W[k * 32 + 1 * 16 + lrow] * mask;   // B row K=1
    v8f acc = {};
    acc = __builtin_amdgcn_wmma_f32_16x16x4_f32(
        false, a, false, b, (short)0, acc, false, false);
    if (full) {
#pragma unroll
      for (int r = 0; r < 8; ++r)
        XW[(long)(tile * 16 + r + 8 * half) * 400 + k * 16 + lrow] = acc[r];
    } else {
#pragma unroll
      for (int r = 0; r < 8; ++r) {
        const int row = tile * 16 + r + 8 * half;
        if (row < N) XW[(long)row * 400 + k * 16 + lrow] = acc[r];
      }
    }
  }
}

// ---------------------------------------------------------------------------
// Per-edge message + scatter: 16 threads per edge (one per out channel).
// Gathers are coalesced 64B rows of XW[src]; XW (160MB) is L2-resident.
// ---------------------------------------------------------------------------
__global__ void edge_kernel(const float* __restrict__ XW,   // [N][25][16]
                            const int* __restrict__ src,
                            const int* __restrict__ dst,
                            const float* __restrict__ ea,   // [E][2]
                            float* __restrict__ agg,        // [N][16]
                            int E) {
  long t = (long)blockIdx.x * blockDim.x + threadIdx.x;
  int e = (int)(t >> 4);
  int j = (int)(t & 15);
  if (e >= E) return;

  const float v0 = ea[2 * e + 0] * 4.0f;     // (K-1) = 4
  const float v1 = ea[2 * e + 1] * 4.0f;
  const float k0 = fminf(fmaxf(floorf(v0), 0.0f), 3.0f);
  const float k1 = fminf(fmaxf(floorf(v1), 0.0f), 3.0f);
  const float f0 = v0 - k0, f1 = v1 - k1;
  const int i0 = (int)k0, i1 = (int)k1;

  const float w00 = (1.0f - f0) * (1.0f - f1);  // (s0,s1)=(0,0) -> wi = p0
  const float w01 = (1.0f - f0) * f1;           // (0,1) -> p0+5
  const float w10 = f0 * (1.0f - f1);           // (1,0) -> p0+1
  const float w11 = f0 * f1;                    // (1,1) -> p0+6

  const long base = (long)src[e] * 400;
  const int p0 = i0 + 5 * i1;
  float m = w00 * XW[base + (p0 + 0) * 16 + j]
          + w01 * XW[base + (p0 + 5) * 16 + j]
          + w10 * XW[base + (p0 + 1) * 16 + j]
          + w11 * XW[base + (p0 + 6) * 16 + j];

  atomicAdd(&agg[(long)dst[e] * 16 + j], m);
}

// ---------------------------------------------------------------------------
// Node update: mean-agg + h@root + bias, ReLU.
// ---------------------------------------------------------------------------
__global__ void node_kernel(const float* __restrict__ agg,   // [N][16]
                            const float* __restrict__ deg,   // [N]
                            const float* __restrict__ h_in,  // [N][IN]
                            const float* __restrict__ root,  // [IN][16]
                            const float* __restrict__ bias,  // [16]
                            float* __restrict__ h_out,       // [N][16]
                            int N, int IN) {
  long t = (long)blockIdx.x * blockDim.x + threadIdx.x;
  int n = (int)(t >> 4);
  int j = (int)(t & 15);
  if (n >= N) return;
  const float d = fmaxf(deg[n], 1.0f);
  float s = agg[(long)n * 16 + j] / d;
  for (int i = 0; i < IN; ++i)
    s += h_in[(long)n * IN + i] * root[i * 16 + j];
  s += bias[j];
  h_out[(long)n * 16 + j] = fmaxf(s, 0.0f);
}

// ---------------------------------------------------------------------------
// Head: logits = h @ fc_w + fc_b ; sigmoid.
// ---------------------------------------------------------------------------
__global__ void final_kernel(const float* __restrict__ h,    // [N][16]
                             const float* __restrict__ fcw,  // [16]
                             const float* __restrict__ fcb,  // [1]
                             float* __restrict__ out, int N) {
  int n = blockIdx.x * blockDim.x + threadIdx.x;
  if (n >= N) return;
  float s = fcb[0];
#pragma unroll
  for (int j = 0; j < 16; ++j) s += h[(long)n * 16 + j] * fcw[j];
  out[n] = 1.0f / (1.0f + expf(-s));
}

// ---------------------------------------------------------------------------
// Launch: degree once; 3 x (WMMA GEMM -> edge scatter -> node update); head.
// Workspace (floats): XW N*400 | agg N*16 | deg N | hA N*16 | hB N*16
//   = N*449 floats ~= 180 MB for N=100k.
// ---------------------------------------------------------------------------
extern "C" void kernel_launch(void* const* d_in, const int* in_sizes, int n_in,
                              void* d_out, int out_size, void* d_ws, size_t ws_size,
                              hipStream_t stream) {
  const float* x     = (const float*)d_in[0];
  const int*   eidx  = (const int*)  d_in[1];
  const float* ea    = (const float*)d_in[2];
  const float* W1    = (const float*)d_in[3];
  const float* root1 = (const float*)d_in[4];
  const float* b1    = (const float*)d_in[5];
  const float* W2    = (const float*)d_in[6];
  const float* root2 = (const float*)d_in[7];
  const float* b2    = (const float*)d_in[8];
  const float* W3    = (const float*)d_in[9];
  const float* root3 = (const float*)d_in[10];
  const float* b3    = (const float*)d_in[11];
  const float* fcw   = (const float*)d_in[12];
  const float* fcb   = (const float*)d_in[13];
  float* out = (float*)d_out;

  const int N = in_sizes[0] / 2;   // x is [N,2]
  const int E = in_sizes[2] / 2;   // edge_attr is [E,2]
  const int* src = eidx;
  const int* dst = eidx + E;

  float* ws  = (float*)d_ws;
  float* XW  = ws;                       // N*400
  float* agg = XW  + (size_t)N * 400;    // N*16
  float* deg = agg + (size_t)N * 16;     // N
  float* hA  = deg + (size_t)N;          // N*16
  float* hB  = hA  + (size_t)N * 16;     // N*16

  const dim3 blk(256);
  const int ntile = (N + 15) / 16;
  const dim3 gXW((ntile + 7) / 8);                       // 8 waves/block
  const dim3 gE((unsigned)(((long)E * 16 + 255) / 256)); // edge*channel
  const dim3 gNode((unsigned)(((long)N * 16 + 255) / 256));
  const dim3 gN((N + 255) / 256);
  const dim3 gEc((E + 255) / 256);

  // degree (fixed across layers)
  zero_f32<<<gN, blk, 0, stream>>>(deg, (long)N);
  degree_kernel<<<gEc, blk, 0, stream>>>(dst, deg, E);

  // ---- layer 1 (in=2) ----
  xw2_wmma<<<gXW, blk, 0, stream>>>(x, W1, XW, N);
  zero_f32<<<gNode, blk, 0, stream>>>(agg, (long)N * 16);
  edge_kernel<<<gE, blk, 0, stream>>>(XW, src, dst, ea, agg, E);
  node_kernel<<<gNode, blk, 0, stream>>>(agg, deg, x, root1, b1, hA, N, 2);

  // ---- layer 2 (in=16) ----
  xw16_wmma<<<gXW, blk, 0, stream>>>(hA, W2, XW, N);
  zero_f32<<<gNode, blk, 0, stream>>>(agg, (long)N * 16);
  edge_kernel<<<gE, blk, 0, stream>>>(XW, src, dst, ea, agg, E);
  node_kernel<<<gNode, blk, 0, stream>>>(agg, deg, hA, root2, b2, hB, N, 16);

  // ---- layer 3 (in=16) ----
  xw16_wmma<<<gXW, blk, 0, stream>>>(hB, W3, XW, N);
  zero_f32<<<gNode, blk, 0, stream>>>(agg, (long)N * 16);
  edge_kernel<<<gE, blk, 0, stream>>>(XW, src, dst, ea, agg, E);
  node_kernel<<<gNode, blk, 0, stream>>>(agg, deg, hB, root3, b3, hA, N, 16);

  // ---- head ----
  final_kernel<<<gN, blk, 0, stream>>>(hA, fcw, fcb, out, N);
}